// Transformer_19413252178467
// MI455X (gfx1250) — compile-verified
//
#include <hip/hip_runtime.h>

// Problem constants (match reference)
#define BB 256
#define TT 256
#define EE 64
#define HH 8
#define HS 8
#define LL 6
#define VV 65
#define FF 256          // 4*E
#define MM (BB*TT)      // 65536 rows

typedef __attribute__((ext_vector_type(2))) float v2f;
typedef __attribute__((ext_vector_type(8))) float v8f;
typedef __attribute__((ext_vector_type(4))) int   v4i;

// Async global->LDS copy path (gfx1250): guard on builtin availability so the
// kernel always compiles; histogram shows whether the async path lowered.
#if defined(__has_builtin)
# if __has_builtin(__builtin_amdgcn_global_load_async_to_lds_b128) && \
     __has_builtin(__builtin_amdgcn_s_wait_asynccnt)
#  define USE_ASYNC_LDS 1
# endif
#endif
#ifndef USE_ASYNC_LDS
# define USE_ASYNC_LDS 0
#endif

#if USE_ASYNC_LDS
// Signature (per clang diagnostics): (v4i AS1*, v4i AS3*, imm int, imm int)
typedef __attribute__((address_space(1))) v4i GV4;   // global int4
typedef __attribute__((address_space(3))) v4i LV4;   // LDS int4
#endif

// ---------------------------------------------------------------------------
// 1) Embedding + sinusoidal positional encoding
// ---------------------------------------------------------------------------
__global__ void embed_kernel(const int* __restrict__ inputs,
                             const float* __restrict__ emb,
                             float* __restrict__ x) {
    int idx = blockIdx.x * blockDim.x + threadIdx.x;   // < MM*EE
    int e   = idx & (EE - 1);
    int row = idx >> 6;                                // b*T + t
    int t   = row & (TT - 1);
    int tok = inputs[row];
    float i2  = (float)((e >> 1) << 1);
    float div = __expf(-(logf(10000.0f) / (float)EE) * i2);
    float ang = (float)t * div;
    float pe  = (e & 1) ? __cosf(ang) : __sinf(ang);
    x[idx] = emb[tok * EE + e] + pe;
}

// ---------------------------------------------------------------------------
// 2) FP32 WMMA GEMM:  C[M,N] = A[M,K] @ W[K,N] + bias  (optional ReLU)
//    Each wave32 owns FOUR 16x16 C tiles stacked in M (64 rows x 16 cols) so
//    every B fragment load feeds 4 back-to-back v_wmma_f32_16x16x4_f32 ops.
// ---------------------------------------------------------------------------
template <int KDIM, int NDIM, bool RELU>
__global__ void gemm_wmma_kernel(const float* __restrict__ A,
                                 const float* __restrict__ W,
                                 const float* __restrict__ bias,
                                 float* __restrict__ C) {
    const int lane  = threadIdx.x & 31;
    const int wave  = threadIdx.x >> 5;
    const int wgl   = blockIdx.x * (blockDim.x >> 5) + wave;
    const int ntile = NDIM / 16;
    const int mb    = wgl / ntile;    // 64-row super-tile index
    const int nt    = wgl % ntile;
    const int half  = lane >> 4;      // 0: K pair {0,1} / rows +0..7 ; 1: {2,3} / rows +8..15
    const int l16   = lane & 15;
    const int mbase = mb * 64;
    const int n     = nt * 16 + l16;  // B/C column owned by this lane

    v8f c0 = {}, c1 = {}, c2 = {}, c3 = {};
    const float* Arow = A + (size_t)(mbase + l16) * KDIM;

    for (int k0 = 0; k0 < KDIM; k0 += 4) {
        const int ka = k0 + half * 2;
        // B 4x16 fragment (shared by all 4 M-tiles): VGPR0=row ka, VGPR1=row ka+1
        v2f b;
        b.x = W[ka * NDIM + n];
        b.y = W[(ka + 1) * NDIM + n];
        // A 16x4 fragments for 4 stacked M-tiles (8B-aligned -> b64 loads)
        v2f a0 = *(const v2f*)(Arow + 0 * 16 * KDIM + ka);
        v2f a1 = *(const v2f*)(Arow + 1 * 16 * KDIM + ka);
        v2f a2 = *(const v2f*)(Arow + 2 * 16 * KDIM + ka);
        v2f a3 = *(const v2f*)(Arow + 3 * 16 * KDIM + ka);
        c0 = __builtin_amdgcn_wmma_f32_16x16x4_f32(false, a0, false, b, (short)0, c0, false, false);
        c1 = __builtin_amdgcn_wmma_f32_16x16x4_f32(false, a1, false, b, (short)0, c1, false, false);
        c2 = __builtin_amdgcn_wmma_f32_16x16x4_f32(false, a2, false, b, (short)0, c2, false, false);
        c3 = __builtin_amdgcn_wmma_f32_16x16x4_f32(false, a3, false, b, (short)0, c3, false, false);
    }

    const float bcol = bias[n];
#pragma unroll
    for (int i = 0; i < 4; ++i) {
        v8f c = (i == 0) ? c0 : (i == 1) ? c1 : (i == 2) ? c2 : c3;
#pragma unroll
        for (int r = 0; r < 8; ++r) {
            int row = mbase + i * 16 + r + half * 8;
            float v = c[r] + bcol;
            if (RELU) v = fmaxf(v, 0.0f);
            C[(size_t)row * NDIM + n] = v;
        }
    }
}

// ---------------------------------------------------------------------------
// 3) Causal attention, one block per (b,h); K/V staged in LDS (16 KB) via
//    async global->LDS copies when available.
// ---------------------------------------------------------------------------
__global__ void attention_kernel(const float* __restrict__ q,
                                 const float* __restrict__ k,
                                 const float* __restrict__ v,
                                 float* __restrict__ o) {
    __shared__ __align__(16) float kbuf[TT][HS];
    __shared__ __align__(16) float vbuf[TT][HS];
    const int bh = blockIdx.x;
    const int b  = bh / HH;
    const int h  = bh % HH;
    const int t  = threadIdx.x;                 // 0..255 = key row / query row

    const float* kp = k + (size_t)(b * TT + t) * EE + h * HS;
    const float* vp = v + (size_t)(b * TT + t) * EE + h * HS;
#if USE_ASYNC_LDS
    __builtin_amdgcn_global_load_async_to_lds_b128((GV4*)kp,       (LV4*)&kbuf[t][0], 0, 0);
    __builtin_amdgcn_global_load_async_to_lds_b128((GV4*)(kp + 4), (LV4*)&kbuf[t][4], 0, 0);
    __builtin_amdgcn_global_load_async_to_lds_b128((GV4*)vp,       (LV4*)&vbuf[t][0], 0, 0);
    __builtin_amdgcn_global_load_async_to_lds_b128((GV4*)(vp + 4), (LV4*)&vbuf[t][4], 0, 0);
    __builtin_amdgcn_s_wait_asynccnt(0);
#else
#pragma unroll
    for (int d = 0; d < HS; ++d) { kbuf[t][d] = kp[d]; vbuf[t][d] = vp[d]; }
#endif
    __syncthreads();

    float qv[HS];
    const float* qp = q + (size_t)(b * TT + t) * EE + h * HS;
#pragma unroll
    for (int d = 0; d < HS; ++d) qv[d] = qp[d];

    const float scale = 0.35355339059327373f;   // 1/sqrt(8)
    float mx = -3.402823e38f;
    for (int j = 0; j <= t; ++j) {
        float s = 0.f;
#pragma unroll
        for (int d = 0; d < HS; ++d) s += qv[d] * kbuf[j][d];
        mx = fmaxf(mx, s * scale);
    }
    float sum = 0.f;
    float acc[HS];
#pragma unroll
    for (int d = 0; d < HS; ++d) acc[d] = 0.f;
    for (int j = 0; j <= t; ++j) {
        float s = 0.f;
#pragma unroll
        for (int d = 0; d < HS; ++d) s += qv[d] * kbuf[j][d];
        float w = __expf(s * scale - mx);
        sum += w;
#pragma unroll
        for (int d = 0; d < HS; ++d) acc[d] += w * vbuf[j][d];
    }
    float inv = 1.f / sum;
    float* op = o + (size_t)(b * TT + t) * EE + h * HS;
#pragma unroll
    for (int d = 0; d < HS; ++d) op[d] = acc[d] * inv;
}

// ---------------------------------------------------------------------------
// 4) x = LayerNorm(x + delta) ; one wave32 per 64-element row
// ---------------------------------------------------------------------------
__global__ void add_ln_kernel(float* __restrict__ x,
                              const float* __restrict__ delta,
                              const float* __restrict__ g,
                              const float* __restrict__ bta) {
    const int wave = threadIdx.x >> 5;
    const int lane = threadIdx.x & 31;
    const size_t row = (size_t)blockIdx.x * 8 + wave;
    const size_t base = row * EE + lane * 2;
    float v0 = x[base] + delta[base];
    float v1 = x[base + 1] + delta[base + 1];
    float s  = v0 + v1;
    float s2 = v0 * v0 + v1 * v1;
#pragma unroll
    for (int off = 16; off > 0; off >>= 1) {
        s  += __shfl_xor(s, off);
        s2 += __shfl_xor(s2, off);
    }
    float mu  = s * (1.0f / EE);
    float var = s2 * (1.0f / EE) - mu * mu;
    float r   = rsqrtf(var + 1e-5f);
    x[base]     = (v0 - mu) * r * g[lane * 2]     + bta[lane * 2];
    x[base + 1] = (v1 - mu) * r * g[lane * 2 + 1] + bta[lane * 2 + 1];
}

// ---------------------------------------------------------------------------
// 5) Output head: logits + log-softmax cross-entropy accumulation
//    One wave32 per row; lane c-loop covers cols {lane, lane+32, lane+64}
// ---------------------------------------------------------------------------
__global__ void logits_loss_kernel(const float* __restrict__ x,
                                   const float* __restrict__ Wout,
                                   const float* __restrict__ bout,
                                   const int* __restrict__ labels,
                                   float* __restrict__ logits,
                                   float* __restrict__ loss_acc) {
    __shared__ float xs[8][EE];
    const int wave = threadIdx.x >> 5;
    const int lane = threadIdx.x & 31;
    const size_t row = (size_t)blockIdx.x * 8 + wave;
    xs[wave][lane * 2]     = x[row * EE + lane * 2];
    xs[wave][lane * 2 + 1] = x[row * EE + lane * 2 + 1];
    __syncthreads();

    float lg[3];
#pragma unroll
    for (int c = 0; c < 3; ++c) {
        int col = lane + 32 * c;
        float acc = -3.402823e38f;
        if (col < VV) {
            acc = bout[col];
            for (int e = 0; e < EE; ++e) acc += xs[wave][e] * Wout[e * VV + col];
            logits[row * VV + col] = acc;
        }
        lg[c] = acc;
    }
    float mx = fmaxf(fmaxf(lg[0], lg[1]), lg[2]);
#pragma unroll
    for (int off = 16; off > 0; off >>= 1) mx = fmaxf(mx, __shfl_xor(mx, off));
    float se = 0.f;
#pragma unroll
    for (int c = 0; c < 3; ++c) {
        int col = lane + 32 * c;
        if (col < VV) se += __expf(lg[c] - mx);
    }
#pragma unroll
    for (int off = 16; off > 0; off >>= 1) se += __shfl_xor(se, off);
    float lse = mx + __logf(se);

    int lbl = labels[row];
    int src_lane = lbl & 31;
    float l0 = __shfl(lg[0], src_lane);
    float l1 = __shfl(lg[1], src_lane);
    float lbl_logit = (lbl < 32) ? l0 : ((lbl < 64) ? l1 : __shfl(lg[2], src_lane));
    if (lane == 0) atomicAdd(loss_acc, lse - lbl_logit);
}

__global__ void zero_kernel(float* p) { p[0] = 0.f; }
__global__ void finalize_kernel(const float* acc, float* out) {
    out[0] = acc[0] * (1.0f / (float)MM);
}

// ---------------------------------------------------------------------------
// Host launcher
// ---------------------------------------------------------------------------
extern "C" void kernel_launch(void* const* d_in, const int* in_sizes, int n_in,
                              void* d_out, int out_size, void* d_ws, size_t ws_size,
                              hipStream_t stream) {
    const int*   inputs = (const int*)  d_in[0];
    const int*   labels = (const int*)  d_in[1];
    const float* emb    = (const float*)d_in[2];
    const float* Wq     = (const float*)d_in[3];
    const float* bq     = (const float*)d_in[4];
    const float* Wk     = (const float*)d_in[5];
    const float* bk     = (const float*)d_in[6];
    const float* Wv     = (const float*)d_in[7];
    const float* bv     = (const float*)d_in[8];
    const float* Wo     = (const float*)d_in[9];
    const float* bo     = (const float*)d_in[10];
    const float* ln1_g  = (const float*)d_in[11];
    const float* ln1_b  = (const float*)d_in[12];
    const float* ln2_g  = (const float*)d_in[13];
    const float* ln2_b  = (const float*)d_in[14];
    const float* W1     = (const float*)d_in[15];
    const float* b1     = (const float*)d_in[16];
    const float* W2     = (const float*)d_in[17];
    const float* b2     = (const float*)d_in[18];
    const float* Wout   = (const float*)d_in[19];
    const float* bout   = (const float*)d_in[20];

    const size_t S = (size_t)MM * EE;       // 4.19M floats per activation buffer
    float* ws     = (float*)d_ws;
    float* x      = ws;                     // [0,S)
    float* qb     = ws + 1 * S;             // [S,2S)
    float* kb     = ws + 2 * S;             // [2S,3S)
    float* vb     = ws + 3 * S;             // [3S,4S)
    float* ao     = ws + 4 * S;             // [4S,5S)
    float* proj   = ws + 5 * S;             // [5S,6S)
    float* hidden = ws + 1 * S;             // reuses q/k/v/ao region: M*FF floats
    float* acc    = ws + 6 * S;             // 1 float

    float* logits   = (float*)d_out;
    float* loss_out = (float*)d_out + (size_t)MM * VV;

    // 1. embedding + positional encoding
    embed_kernel<<<(MM * EE) / 256, 256, 0, stream>>>(inputs, emb, x);

    // Waves per GEMM = (M/64) * (N/16); 8 waves per 256-thread block.
    const int gemm64  = (MM / 64) * (EE / 16) / 8;   //  512 blocks (N=64)
    const int gemm256 = (MM / 64) * (FF / 16) / 8;   // 2048 blocks (N=256)

    for (int l = 0; l < LL; ++l) {
        const float* Wql = Wq + (size_t)l * EE * EE;
        const float* Wkl = Wk + (size_t)l * EE * EE;
        const float* Wvl = Wv + (size_t)l * EE * EE;
        const float* Wol = Wo + (size_t)l * EE * EE;
        const float* W1l = W1 + (size_t)l * EE * FF;
        const float* W2l = W2 + (size_t)l * FF * EE;

        gemm_wmma_kernel<EE, EE, false><<<gemm64, 256, 0, stream>>>(x, Wql, bq + l * EE, qb);
        gemm_wmma_kernel<EE, EE, false><<<gemm64, 256, 0, stream>>>(x, Wkl, bk + l * EE, kb);
        gemm_wmma_kernel<EE, EE, false><<<gemm64, 256, 0, stream>>>(x, Wvl, bv + l * EE, vb);

        attention_kernel<<<BB * HH, TT, 0, stream>>>(qb, kb, vb, ao);

        gemm_wmma_kernel<EE, EE, false><<<gemm64, 256, 0, stream>>>(ao, Wol, bo + l * EE, proj);
        add_ln_kernel<<<MM / 8, 256, 0, stream>>>(x, proj, ln1_g + l * EE, ln1_b + l * EE);

        gemm_wmma_kernel<EE, FF, true ><<<gemm256, 256, 0, stream>>>(x, W1l, b1 + l * FF, hidden);
        gemm_wmma_kernel<FF, EE, false><<<gemm64, 256, 0, stream>>>(hidden, W2l, b2 + l * EE, proj);
        add_ln_kernel<<<MM / 8, 256, 0, stream>>>(x, proj, ln2_g + l * EE, ln2_b + l * EE);
    }

    zero_kernel<<<1, 1, 0, stream>>>(acc);
    logits_loss_kernel<<<MM / 8, 256, 0, stream>>>(x, Wout, bout, labels, logits, acc);
    finalize_kernel<<<1, 1, 0, stream>>>(acc, loss_out);
}